// SEGCL_72868415144396
// MI455X (gfx1250) — compile-verified
//
#include <hip/hip_runtime.h>
#include <math.h>

typedef __attribute__((ext_vector_type(2))) float v2f;
typedef __attribute__((ext_vector_type(8))) float v8f;

#define HID 128
#define EMB 16

// ---------------- elementwise helpers ----------------
__global__ void zero_kernel(float* __restrict__ p, int n) {
    int i = blockIdx.x * blockDim.x + threadIdx.x;
    if (i < n) p[i] = 0.0f;
}

__global__ void copy_kernel(const float* __restrict__ in, float* __restrict__ out, int n) {
    int i = blockIdx.x * blockDim.x + threadIdx.x;
    if (i < n) out[i] = in[i];
}

// Tx2 = 2*Tx2 - Tx0
__global__ void axpby_kernel(const float* __restrict__ Tx0, float* __restrict__ Tx2, int n) {
    int i = blockIdx.x * blockDim.x + threadIdx.x;
    if (i < n) Tx2[i] = 2.0f * Tx2[i] - Tx0[i];
}

// ---------------- graph preprocessing ----------------
__global__ void degree_kernel(const int* __restrict__ src, float* __restrict__ deg, int E) {
    int e = blockIdx.x * blockDim.x + threadIdx.x;
    if (e < E) atomicAdd(deg + src[e], 1.0f);
}

__global__ void dinv_kernel(float* __restrict__ deg, int N) {
    int i = blockIdx.x * blockDim.x + threadIdx.x;
    if (i < N) {
        float d = deg[i];
        deg[i] = (d > 0.0f) ? rsqrtf(fmaxf(d, 1.0f)) : 0.0f;
    }
}

__global__ void norm_kernel(const int* __restrict__ src, const int* __restrict__ dst,
                            const float* __restrict__ dinv, float* __restrict__ nrm, int E) {
    int e = blockIdx.x * blockDim.x + threadIdx.x;
    if (e < E) nrm[e] = -dinv[src[e]] * dinv[dst[e]];
}

// ---------------- feature assembly: x = [id | len | lng | lat | vis] ----------------
// one wave32 per node, each lane writes 4 of the 128 features
__global__ void gather_kernel(const int* __restrict__ attr, const float* __restrict__ vis,
                              const float* __restrict__ e_len, const float* __restrict__ e_id,
                              const float* __restrict__ e_lng, const float* __restrict__ e_lat,
                              float* __restrict__ x, int N) {
    int gid = blockIdx.x * blockDim.x + threadIdx.x;
    int n = gid >> 5;
    int lane = gid & 31;
    if (n >= N) return;
    int f = lane * 4;
    float4 v;
    if (f < 64) {
        const float* tab;
        int idx;
        switch (f >> 4) {
            case 0:  tab = e_id;  idx = attr[n * 4 + 1]; break;
            case 1:  tab = e_len; idx = attr[n * 4 + 0]; break;
            case 2:  tab = e_lng; idx = attr[n * 4 + 2]; break;
            default: tab = e_lat; idx = attr[n * 4 + 3]; break;
        }
        v = *(const float4*)(tab + (size_t)idx * EMB + (f & 15));
    } else {
        v = *(const float4*)(vis + (size_t)n * 64 + (f - 64));
    }
    *(float4*)(x + (size_t)n * HID + f) = v;
}

// ---------------- sparse propagation: out[dst] += norm * t[src] ----------------
// one wave32 per edge; lane moves float4 (128 features / 32 lanes).
// src/dst/norm are wave-uniform -> force into SGPRs via readfirstlane so the
// compiler issues scalar loads instead of 32 redundant per-lane VMEM loads.
__global__ void prop_kernel(const float* __restrict__ tin, const float* __restrict__ nrm,
                            const int* __restrict__ src, const int* __restrict__ dst,
                            float* __restrict__ tout, int E) {
    int gid = blockIdx.x * blockDim.x + threadIdx.x;
    int e = __builtin_amdgcn_readfirstlane(gid >> 5);
    int lane = gid & 31;
    if (e >= E) return;
    int s = __builtin_amdgcn_readfirstlane(src[e]);
    int d = __builtin_amdgcn_readfirstlane(dst[e]);
    float w = nrm[e];
    float4 v = *(const float4*)(tin + (size_t)s * HID + lane * 4);
    float* o = tout + (size_t)d * HID + lane * 4;
    atomicAdd(o + 0, w * v.x);
    atomicAdd(o + 1, w * v.y);
    atomicAdd(o + 2, w * v.z);
    atomicAdd(o + 3, w * v.w);
}

// ---------------- fp32 WMMA GEMM ----------------
// out[M,NOUT] = act( [A0|A1|A2][M, 128*NSEG] @ W[128*NSEG, NOUT] + bias )
// A segments are [M,128] row-major; W row-major. Grid: ceil(M/16) blocks;
// blockDim = 32*(NOUT/16); each wave owns one 16x16 tile.
// ACT: 0 = none, 1 = relu, 2 = elu. All shape params compile-time so the
// K-loop unrolls and loads pipeline ahead of the v_wmma chain.
template <int NSEG, int ACT, int NOUT>
__global__ __launch_bounds__(256) void wmma_gemm_kernel(
        const float* __restrict__ A0, const float* __restrict__ A1,
        const float* __restrict__ A2, const float* __restrict__ W,
        const float* __restrict__ bias, float* __restrict__ out, int M) {
    constexpr int K = 128 * NSEG;
    constexpr int KP = K + 4;  // padded row stride -> conflict-free ds reads
    __shared__ __align__(16) float sA[16 * KP];
    const int row0 = blockIdx.x * 16;

    // cooperative stage of the 16 x K A-slab into LDS (float4, fully coalesced)
    {
        const float* segs[3] = {A0, A1, A2};
        constexpr int nf4 = (16 * K) >> 2;
        for (int idx = threadIdx.x; idx < nf4; idx += blockDim.x) {
            int r = idx / (K >> 2);
            int k = (idx % (K >> 2)) << 2;
            int row = row0 + r;
            if (row >= M) row = M - 1;  // safety clamp for a ragged last tile
            const float* seg = (NSEG == 1) ? A0 : segs[k >> 7];
            float4 v = *(const float4*)(seg + (size_t)row * 128 + (k & 127));
            *(float4*)(sA + r * KP + k) = v;
        }
    }
    __syncthreads();

    const int lane = threadIdx.x & 31;
    const int ln = lane & 15;
    const int hi = lane >> 4;       // 0 -> K+0,1 / rows 0-7 ; 1 -> K+2,3 / rows 8-15
    const int koff = hi << 1;
    const int col = (threadIdx.x >> 5) * 16 + ln;

    const float* ap = sA + ln * KP + koff;         // A pair is contiguous -> ds_load_b64
    const float* wp = W + (size_t)koff * NOUT + col;

    v8f acc = {};
#pragma unroll 8
    for (int k0 = 0; k0 < K; k0 += 4) {
        v2f a = *(const v2f*)ap;
        v2f b;
        b.x = wp[0];
        b.y = wp[NOUT];
        acc = __builtin_amdgcn_wmma_f32_16x16x4_f32(false, a, false, b,
                                                    (short)0, acc, false, false);
        ap += 4;
        wp += 4 * NOUT;
    }

    const float bv = bias[col];
    const int rbase = row0 + (hi << 3);
    float* op = out + (size_t)rbase * NOUT + col;
    if (row0 + 16 <= M) {
        // full tile: branch-free epilogue (common case; N is a multiple of 16)
#pragma unroll
        for (int i = 0; i < 8; ++i) {
            float v = acc[i] + bv;
            if (ACT == 1)      v = fmaxf(v, 0.0f);
            else if (ACT == 2) v = (v > 0.0f) ? v : (expf(v) - 1.0f);
            op[(size_t)i * NOUT] = v;
        }
    } else {
#pragma unroll
        for (int i = 0; i < 8; ++i) {
            if (rbase + i < M) {
                float v = acc[i] + bv;
                if (ACT == 1)      v = fmaxf(v, 0.0f);
                else if (ACT == 2) v = (v > 0.0f) ? v : (expf(v) - 1.0f);
                op[(size_t)i * NOUT] = v;
            }
        }
    }
}

// ---------------- driver ----------------
extern "C" void kernel_launch(void* const* d_in, const int* in_sizes, int n_in,
                              void* d_out, int out_size, void* d_ws, size_t ws_size,
                              hipStream_t stream) {
    const int*   edge_index = (const int*)d_in[0];
    const int*   node_attr  = (const int*)d_in[1];
    const float* vis        = (const float*)d_in[2];
    const float* e_len      = (const float*)d_in[3];
    const float* e_id       = (const float*)d_in[4];
    const float* e_lng      = (const float*)d_in[5];
    const float* e_lat      = (const float*)d_in[6];
    const float* W0  = (const float*)d_in[7];
    const float* b0  = (const float*)d_in[8];
    const float* W1  = (const float*)d_in[9];
    const float* b1  = (const float*)d_in[10];
    const float* P1  = (const float*)d_in[11];
    const float* pb1 = (const float*)d_in[12];
    const float* P2  = (const float*)d_in[13];
    const float* pb2 = (const float*)d_in[14];

    const int E = in_sizes[0] / 2;
    const int N = in_sizes[1] / 4;
    const int* src = edge_index;
    const int* dst = edge_index + E;

    // workspace layout (floats): Tx0 | Tx1 | Tx2 | deg | norm  (~80 MB)
    float* Tx0 = (float*)d_ws;
    float* Tx1 = Tx0 + (size_t)N * HID;
    float* Tx2 = Tx1 + (size_t)N * HID;
    float* deg = Tx2 + (size_t)N * HID;
    float* nrm = deg + N;

    float* h = (float*)d_out;            // [N,128]
    float* z = h + (size_t)N * HID;      // [N,64]

    const int TB = 256;
    auto nb = [](long n, int tb) { return (int)((n + tb - 1) / tb); };
    const int NH = N * HID;
    const int mtiles = (N + 15) / 16;

    // graph preprocessing
    zero_kernel  <<<nb(N, TB), TB, 0, stream>>>(deg, N);
    degree_kernel<<<nb(E, TB), TB, 0, stream>>>(src, deg, E);
    dinv_kernel  <<<nb(N, TB), TB, 0, stream>>>(deg, N);
    norm_kernel  <<<nb(E, TB), TB, 0, stream>>>(src, dst, deg, nrm, E);

    // x -> Tx0
    gather_kernel<<<nb((long)N * 32, TB), TB, 0, stream>>>(node_attr, vis, e_len, e_id,
                                                           e_lng, e_lat, Tx0, N);

    // two ChebConv(K=3) + ReLU layers
    for (int layer = 0; layer < 2; ++layer) {
        const float* W = layer ? W1 : W0;   // [3,128,128] == [384,128] stacked over K
        const float* b = layer ? b1 : b0;
        if (layer)
            copy_kernel<<<nb(NH, TB), TB, 0, stream>>>(h, Tx0, NH);
        zero_kernel<<<nb(NH, TB), TB, 0, stream>>>(Tx1, NH);
        zero_kernel<<<nb(NH, TB), TB, 0, stream>>>(Tx2, NH);
        prop_kernel<<<nb((long)E * 32, TB), TB, 0, stream>>>(Tx0, nrm, src, dst, Tx1, E);
        prop_kernel<<<nb((long)E * 32, TB), TB, 0, stream>>>(Tx1, nrm, src, dst, Tx2, E);
        axpby_kernel<<<nb(NH, TB), TB, 0, stream>>>(Tx0, Tx2, NH);
        wmma_gemm_kernel<3, 1, 128><<<mtiles, 256, 0, stream>>>(Tx0, Tx1, Tx2, W, b, h, N);
    }

    // projection head: z = elu(h @ P1 + pb1) @ P2 + pb2   (Tx1 recycled as intermediate)
    wmma_gemm_kernel<1, 2, 128><<<mtiles, 256, 0, stream>>>(h, nullptr, nullptr,
                                                            P1, pb1, Tx1, N);
    wmma_gemm_kernel<1, 0, 64><<<mtiles, 128, 0, stream>>>(Tx1, nullptr, nullptr,
                                                           P2, pb2, z, N);
}